// polygnn_mp_83202106458600
// MI455X (gfx1250) — compile-verified
//
#include <hip/hip_runtime.h>
#include <hip/hip_bf16.h>
#include <math.h>

typedef __attribute__((ext_vector_type(2))) float v2f;
typedef __attribute__((ext_vector_type(8))) float v8f;

#define N_NODES    50000
#define N_EDGES    800000
#define NODE_F     64
#define EDGE_F     32
#define UP_F       160      // 2*NODE + EDGE
#define READ_F     256
#define NUM_GRAPHS 512
#define BN_EPS     1e-5f
#define NODE_TILES (N_NODES / 16)   // 3125 exact

__device__ __forceinline__ v8f wmma4(v2f a, v2f b, v8f c) {
  // D = A(16x4 f32) x B(4x16 f32) + C(16x16 f32), full fp32 precision
  return __builtin_amdgcn_wmma_f32_16x16x4_f32(false, a, false, b, (short)0, c,
                                               false, false);
}

__device__ __forceinline__ void atomic_add_f32(float* p, float v) {
  unsafeAtomicAdd(p, v);   // hardware global_atomic_add_f32
}

// Pack W (2P x C, row-major) into pair layout: sWp[r*S + n] = (W[2r][n], W[2r+1][n])
// S > C with (S-C)*8 bytes = 32-bank skew so the two half-wave pair-rows avoid
// LDS bank conflicts on ds_load_b64.
template <int P, int C, int S>
__device__ __forceinline__ void pack_pairs(v2f* __restrict__ sWp,
                                           const float* __restrict__ W, int tid) {
  for (int i = tid; i < P * C; i += 256) {
    int r = i / C, n = i % C;
    v2f w;
    w.x = W[(2 * r) * C + n];
    w.y = W[(2 * r + 1) * C + n];
    sWp[r * S + n] = w;
  }
}

// ---------------------------------------------------------------- utilities
__global__ void zero_kernel(float* __restrict__ p, long n) {
  long i = (long)blockIdx.x * blockDim.x + threadIdx.x;
  if (i < n) p[i] = 0.f;
}

// h_in = a + b   (only needed for depths 2,3)
__global__ void prep_add_kernel(const float* __restrict__ a,
                                const float* __restrict__ b,
                                float* __restrict__ h_in, int n) {
  int i = blockIdx.x * blockDim.x + threadIdx.x;
  if (i < n) h_in[i] = a[i] + b[i];
}

// ------------------------------------------------ one-time edge FFN + scatter
// m_const[dst] += w_bonds * relu(edge_attr @ E_W + E_b)      (loop-invariant)
__global__ __launch_bounds__(256)
void edge_ffn_const_kernel(const float* __restrict__ edge_attr,
                           const int* __restrict__ dst,
                           const float* __restrict__ w_bonds,
                           const float* __restrict__ E_W,
                           const float* __restrict__ E_b,
                           float* __restrict__ m_const) {
  __shared__ v2f   sWp[16 * 48];                     // 6 KB, pair-packed E_W
  __shared__ float sB[EDGE_F];
  int tid = threadIdx.x;
  pack_pairs<16, 32, 48>(sWp, E_W, tid);
  if (tid < EDGE_F) sB[tid] = E_b[tid];
  __syncthreads();

  int wave = tid >> 5, lane = tid & 31;
  int lo = lane & 15, hi = lane >> 4;
  int e0 = blockIdx.x * 128 + wave * 16;             // 16-edge tile
  const v2f* arow = (const v2f*)(edge_attr + (size_t)(e0 + lo) * EDGE_F);

  v8f acc[2] = {};
#pragma unroll
  for (int kk = 0; kk < 8; ++kk) {                   // K = 32
    v2f a = arow[kk * 2 + hi];                       // global_load_b64
#pragma unroll
    for (int nt = 0; nt < 2; ++nt) {                 // N = 32
      v2f b = sWp[(kk * 2 + hi) * 48 + nt * 16 + lo]; // ds_load_b64
      acc[nt] = wmma4(a, b, acc[nt]);
    }
  }
  float bias[2];
#pragma unroll
  for (int nt = 0; nt < 2; ++nt) bias[nt] = sB[nt * 16 + lo];
#pragma unroll
  for (int v = 0; v < 8; ++v) {
    int e = e0 + v + 8 * hi;
    float wbv = w_bonds[e];
    float* p = m_const + (size_t)dst[e] * EDGE_F + lo;
#pragma unroll
    for (int nt = 0; nt < 2; ++nt)
      atomic_add_f32(p + nt * 16, fmaxf(acc[nt][v] + bias[nt], 0.f) * wbv);
  }
}

// -------------------------------------------- per-depth edge message passing
// aggr[dst, 0:64] += w_bonds * relu(h_in[src] @ V_W + V_b)
__global__ __launch_bounds__(256)
void edge_mp_kernel(const float* __restrict__ h_in,
                    const int* __restrict__ src,
                    const int* __restrict__ dst,
                    const float* __restrict__ w_bonds,
                    const float* __restrict__ V_W,
                    const float* __restrict__ V_b,
                    float* __restrict__ aggr) {
  __shared__ v2f   sWp[32 * 80];                     // 20 KB, pair-packed V_W
  __shared__ float sB[NODE_F];
  int tid = threadIdx.x;
  pack_pairs<32, 64, 80>(sWp, V_W, tid);
  if (tid < NODE_F) sB[tid] = V_b[tid];
  __syncthreads();

  int wave = tid >> 5, lane = tid & 31;
  int lo = lane & 15, hi = lane >> 4;
  int e0 = blockIdx.x * 128 + wave * 16;
  int my_src = src[e0 + lo];
  const v2f* arow = (const v2f*)(h_in + (size_t)my_src * NODE_F); // gathered row

  v8f acc[4] = {};
#pragma unroll
  for (int kk = 0; kk < 16; ++kk) {                  // K = 64
    v2f a = arow[kk * 2 + hi];                       // global_load_b64
#pragma unroll
    for (int nt = 0; nt < 4; ++nt) {                 // N = 64
      v2f b = sWp[(kk * 2 + hi) * 80 + nt * 16 + lo]; // ds_load_b64
      acc[nt] = wmma4(a, b, acc[nt]);
    }
  }
  float bias[4];
#pragma unroll
  for (int nt = 0; nt < 4; ++nt) bias[nt] = sB[nt * 16 + lo];
#pragma unroll
  for (int v = 0; v < 8; ++v) {
    int e = e0 + v + 8 * hi;
    float wbv = w_bonds[e];
    float* p = aggr + (size_t)dst[e] * NODE_F + lo;  // one base, const offsets
#pragma unroll
    for (int nt = 0; nt < 4; ++nt)
      atomic_add_f32(p + nt * 16, fmaxf(acc[nt][v] + bias[nt], 0.f) * wbv);
  }
}

// -------------------------------------------------- per-depth node update
// h_out = relu(elu(concat(aggr64, m_const32, h_in64)) @ U_W + U_b) * w_atoms
__device__ __forceinline__ float up_elu(const float* __restrict__ aggr,
                                        const float* __restrict__ m_const,
                                        const float* __restrict__ h_in,
                                        int node, int col) {
  float v;
  if (col < NODE_F)                 v = aggr[(size_t)node * NODE_F + col];
  else if (col < NODE_F + EDGE_F)   v = m_const[(size_t)node * EDGE_F + (col - NODE_F)];
  else                              v = h_in[(size_t)node * NODE_F + (col - NODE_F - EDGE_F)];
  return v > 0.f ? v : expm1f(v);   // elu
}

__global__ __launch_bounds__(256)
void node_update_kernel(const float* __restrict__ aggr,
                        const float* __restrict__ m_const,
                        const float* __restrict__ h_in,
                        const float* __restrict__ U_W,
                        const float* __restrict__ U_b,
                        const float* __restrict__ w_atoms,
                        float* __restrict__ h_out) {
  __shared__ v2f   sWp[80 * 80];                     // 51.2 KB, pair-packed U_W
  __shared__ float sB[NODE_F];
  int tid = threadIdx.x;
  pack_pairs<80, 64, 80>(sWp, U_W, tid);
  if (tid < NODE_F) sB[tid] = U_b[tid];
  __syncthreads();

  int wave = tid >> 5, lane = tid & 31;
  int lo = lane & 15, hi = lane >> 4;
  int t = blockIdx.x * 8 + wave;
  if (t >= NODE_TILES) return;                       // wave-uniform
  int node = t * 16 + lo;

  v8f acc[4] = {};
#pragma unroll 4
  for (int kk = 0; kk < 40; ++kk) {                  // K = 160
    int k = kk * 4 + hi * 2;
    v2f a;
    a.x = up_elu(aggr, m_const, h_in, node, k);
    a.y = up_elu(aggr, m_const, h_in, node, k + 1);
#pragma unroll
    for (int nt = 0; nt < 4; ++nt) {                 // N = 64
      v2f b = sWp[(kk * 2 + hi) * 80 + nt * 16 + lo];
      acc[nt] = wmma4(a, b, acc[nt]);
    }
  }
  float bias[4];
#pragma unroll
  for (int nt = 0; nt < 4; ++nt) bias[nt] = sB[nt * 16 + lo];
#pragma unroll
  for (int v = 0; v < 8; ++v) {
    int nm = t * 16 + v + 8 * hi;
    float wav = w_atoms[nm];
    float* p = h_out + (size_t)nm * NODE_F + lo;
#pragma unroll
    for (int nt = 0; nt < 4; ++nt)
      p[nt * 16] = fmaxf(acc[nt][v] + bias[nt], 0.f) * wav;
  }
}

// ------------------------------------------------------------- readout GEMM
// mode 0: accumulate per-column sum/sumsq of y = (h+x)@R_W + R_b
// mode 1: y_bn = relu(y*scale+shift); segment-sum into gsum[batch]
__global__ __launch_bounds__(256)
void readout_pass_kernel(const float* __restrict__ h,
                         const float* __restrict__ x,
                         const float* __restrict__ R_W,
                         const float* __restrict__ R_b,
                         int mode,
                         const float* __restrict__ scale,
                         const float* __restrict__ shift,
                         const int* __restrict__ batch,
                         float* __restrict__ colsum,
                         float* __restrict__ colsumsq,
                         float* __restrict__ gsum) {
  __shared__ v2f   sWp[32 * 272];                    // 69.6 KB, pair-packed R_W
  __shared__ float sB[READ_F];
  int tid = threadIdx.x;
  pack_pairs<32, 256, 272>(sWp, R_W, tid);
  if (tid < READ_F) sB[tid] = R_b[tid];
  __syncthreads();

  int wave = tid >> 5, lane = tid & 31;
  int lo = lane & 15, hi = lane >> 4;
  int t  = blockIdx.x * 2 + (wave >> 2);             // node tile
  int cg = wave & 3;                                 // 64-col group
  if (t >= NODE_TILES) return;                       // wave-uniform
  int node = t * 16 + lo;
  const v2f* hrow = (const v2f*)(h + (size_t)node * NODE_F);
  const v2f* xrow = (const v2f*)(x + (size_t)node * NODE_F);

  v8f acc[4] = {};
#pragma unroll
  for (int kk = 0; kk < 16; ++kk) {                  // K = 64
    v2f a = hrow[kk * 2 + hi] + xrow[kk * 2 + hi];
#pragma unroll
    for (int nt = 0; nt < 4; ++nt) {
      v2f b = sWp[(kk * 2 + hi) * 272 + cg * 64 + nt * 16 + lo];
      acc[nt] = wmma4(a, b, acc[nt]);
    }
  }

  if (mode == 0) {
#pragma unroll
    for (int nt = 0; nt < 4; ++nt) {
      int c = cg * 64 + nt * 16 + lo;
      float bias = sB[c], s = 0.f, sq = 0.f;
#pragma unroll
      for (int v = 0; v < 8; ++v) {
        float val = acc[nt][v] + bias;
        s += val; sq += val * val;
      }
      s  += __shfl_xor(s, 16, 32);                   // lanes l / l+16 share c
      sq += __shfl_xor(sq, 16, 32);
      if (hi == 0) {
        atomic_add_f32(&colsum[c], s);
        atomic_add_f32(&colsumsq[c], sq);
      }
    }
  } else {
    int g[8];
#pragma unroll
    for (int v = 0; v < 8; ++v) g[v] = batch[t * 16 + v + 8 * hi];
#pragma unroll
    for (int nt = 0; nt < 4; ++nt) {
      int c = cg * 64 + nt * 16 + lo;
      float bias = sB[c], sc = scale[c], sh = shift[c];
      float run = 0.f; int rg = g[0];
#pragma unroll
      for (int v = 0; v < 8; ++v) {                  // batch sorted: merge runs
        float val = fmaxf((acc[nt][v] + bias) * sc + sh, 0.f);
        if (g[v] != rg) {
          atomic_add_f32(&gsum[(size_t)rg * READ_F + c], run);
          rg = g[v]; run = 0.f;
        }
        run += val;
      }
      atomic_add_f32(&gsum[(size_t)rg * READ_F + c], run);
    }
  }
}

__global__ void bn_stats_kernel(const float* __restrict__ colsum,
                                const float* __restrict__ colsumsq,
                                const float* __restrict__ gamma,
                                const float* __restrict__ beta,
                                float* __restrict__ scale,
                                float* __restrict__ shift) {
  int c = threadIdx.x;                               // 256 threads, 1 block
  float inv_n = 1.0f / (float)N_NODES;
  float mu  = colsum[c] * inv_n;
  float var = colsumsq[c] * inv_n - mu * mu;
  float sc  = rsqrtf(var + BN_EPS) * gamma[c];
  scale[c] = sc;
  shift[c] = beta[c] - mu * sc;
}

__global__ void count_kernel(const int* __restrict__ batch, float* __restrict__ cnt) {
  int i = blockIdx.x * blockDim.x + threadIdx.x;
  if (i < N_NODES) atomic_add_f32(&cnt[batch[i]], 1.0f);
}

__global__ void finalize_kernel(const float* __restrict__ gsum,
                                const float* __restrict__ cnt,
                                float* __restrict__ out) {
  int i = blockIdx.x * blockDim.x + threadIdx.x;
  if (i < NUM_GRAPHS * READ_F) out[i] = gsum[i] / fmaxf(cnt[i >> 8], 1.0f);
}

// =========================================================================
extern "C" void kernel_launch(void* const* d_in, const int* in_sizes, int n_in,
                              void* d_out, int out_size, void* d_ws, size_t ws_size,
                              hipStream_t stream) {
  (void)in_sizes; (void)n_in; (void)out_size; (void)ws_size;
  const float* x         = (const float*)d_in[0];
  const int*   ei        = (const int*)  d_in[1];
  const float* edge_attr = (const float*)d_in[2];
  const float* w_atoms   = (const float*)d_in[3];
  const float* w_bonds   = (const float*)d_in[4];
  const int*   batch     = (const int*)  d_in[5];
  const float* V_W = (const float*)d_in[6];
  const float* V_b = (const float*)d_in[7];
  const float* E_W = (const float*)d_in[8];
  const float* E_b = (const float*)d_in[9];
  const float* U_W = (const float*)d_in[10];
  const float* U_b = (const float*)d_in[11];
  const float* R_W = (const float*)d_in[12];
  const float* R_b = (const float*)d_in[13];
  const float* R_gamma = (const float*)d_in[14];
  const float* R_beta  = (const float*)d_in[15];
  const int* src = ei;
  const int* dst = ei + N_EDGES;

  // ---- workspace layout (floats) ----
  float* ws = (float*)d_ws;
  float* m_const = ws;  ws += (size_t)N_NODES * EDGE_F;   // 6.4 MB
  float* aggr    = ws;  ws += (size_t)N_NODES * NODE_F;   // 12.8 MB
  float* h_in    = ws;  ws += (size_t)N_NODES * NODE_F;
  float* hA      = ws;  ws += (size_t)N_NODES * NODE_F;
  float* hB      = ws;  ws += (size_t)N_NODES * NODE_F;
  float* hC      = ws;  ws += (size_t)N_NODES * NODE_F;
  float* colsum   = ws; ws += READ_F;                     // contiguous tail...
  float* colsumsq = ws; ws += READ_F;
  float* bn_scale = ws; ws += READ_F;
  float* bn_shift = ws; ws += READ_F;
  float* gsum     = ws; ws += (size_t)NUM_GRAPHS * READ_F;
  float* cnt      = ws; ws += NUM_GRAPHS;

  // ---- zero the accumulators ----
  {
    long n1 = (long)N_NODES * EDGE_F;
    zero_kernel<<<(int)((n1 + 255) / 256), 256, 0, stream>>>(m_const, n1);
    long n2 = 4 * READ_F + (long)NUM_GRAPHS * READ_F + NUM_GRAPHS;
    zero_kernel<<<(int)((n2 + 255) / 256), 256, 0, stream>>>(colsum, n2);
  }

  // ---- loop-invariant edge FFN + scatter (once) ----
  edge_ffn_const_kernel<<<N_EDGES / 128, 256, 0, stream>>>(
      edge_attr, dst, w_bonds, E_W, E_b, m_const);

  // ---- 4 depths of message passing ----
  // H1=mp(x)->hA, H2=mp(H1)->hB, H3=mp(H1+H2)->hC, H4=mp(H2+H3)->hA
  struct Step { const float* pp; const float* prev; float* out; };
  Step steps[4] = { {nullptr, x,  hA}, {nullptr, hA, hB},
                    {hA,      hB, hC}, {hB,      hC, hA} };
  const int nelem = N_NODES * NODE_F;
  const int nblk  = (nelem + 255) / 256;
  for (int d = 0; d < 4; ++d) {
    zero_kernel<<<nblk, 256, 0, stream>>>(aggr, nelem);
    const float* hin = steps[d].prev;                // depths 0/1: no copy needed
    if (steps[d].pp) {
      prep_add_kernel<<<nblk, 256, 0, stream>>>(steps[d].pp, steps[d].prev,
                                                h_in, nelem);
      hin = h_in;
    }
    edge_mp_kernel<<<N_EDGES / 128, 256, 0, stream>>>(
        hin, src, dst, w_bonds, V_W, V_b, aggr);
    node_update_kernel<<<(NODE_TILES + 7) / 8, 256, 0, stream>>>(
        aggr, m_const, hin, U_W, U_b, w_atoms, steps[d].out);
  }
  const float* h_final = hA;

  // ---- readout: stats pass, BN params, counts, apply+segment-sum, divide ----
  int ro_blocks = (NODE_TILES + 1) / 2;
  readout_pass_kernel<<<ro_blocks, 256, 0, stream>>>(
      h_final, x, R_W, R_b, /*mode=*/0, nullptr, nullptr, nullptr,
      colsum, colsumsq, nullptr);
  bn_stats_kernel<<<1, 256, 0, stream>>>(colsum, colsumsq, R_gamma, R_beta,
                                         bn_scale, bn_shift);
  count_kernel<<<(N_NODES + 255) / 256, 256, 0, stream>>>(batch, cnt);
  readout_pass_kernel<<<ro_blocks, 256, 0, stream>>>(
      h_final, x, R_W, R_b, /*mode=*/1, bn_scale, bn_shift, batch,
      nullptr, nullptr, gsum);
  finalize_kernel<<<(NUM_GRAPHS * READ_F + 255) / 256, 256, 0, stream>>>(
      gsum, cnt, (float*)d_out);
}